// SubjectBlockTest2_79534204387494
// MI455X (gfx1250) — compile-verified
//
#include <hip/hip_runtime.h>
#include <math.h>

typedef __attribute__((ext_vector_type(2))) float v2f;
typedef __attribute__((ext_vector_type(8))) float v8f;
typedef int v4i __attribute__((vector_size(16)));   // matches builtin param type

#define D1V   270
#define CCV   272
#define TTV   1024
#define BBV   128
#define KLT   1024     // K*K harmonics flattened
#define MPAD  288      // 270 padded to 18*16
#define KDIM  272      // GEMM reduction dim (C, and padded 270)
#define MBLK  96       // 6 x 16 rows per workgroup
#define NBLK  128      // 8 x 16 cols per workgroup (one 16-col strip per wave)
#define KCH   16       // K chunk staged in LDS per iteration
#define ASTR  276      // LDS stride for A block (bank-conflict avoidance)

#if defined(__gfx1250__) && __has_builtin(__builtin_amdgcn_global_load_async_to_lds_b128)
#define USE_ASYNC_LDS 1
#else
#define USE_ASYNC_LDS 0
#endif

// Issue an async 16-byte global->LDS copy (no VGPR staging, tracked by ASYNCcnt)
__device__ __forceinline__ void async_copy_b128(const float* gsrc, float* ldst) {
#if USE_ASYNC_LDS
  __builtin_amdgcn_global_load_async_to_lds_b128((v4i*)gsrc, (v4i*)ldst, 0, 0);
#else
  float4 v = *(const float4*)gsrc;
  *(float4*)ldst = v;
#endif
}

__device__ __forceinline__ void async_wait0() {
#if USE_ASYNC_LDS
#if __has_builtin(__builtin_amdgcn_s_wait_asynccnt)
  __builtin_amdgcn_s_wait_asynccnt(0);
#else
  asm volatile("s_wait_asynccnt 0" ::: "memory");
#endif
#endif
}

// ---------------- small prep kernels ----------------

__global__ void k_trig(const float* __restrict__ loc,
                       float* __restrict__ cosT, float* __restrict__ sinT) {
  int idx = blockIdx.x * blockDim.x + threadIdx.x;
  if (idx >= KLT * CCV) return;
  int kl = idx / CCV, c = idx - kl * CCV;
  float kh = (float)(kl >> 5);
  float lh = (float)(kl & 31);
  float ph = 6.283185307179586f * (kh * loc[c * 2 + 0] + lh * loc[c * 2 + 1]);
  float s, co;
  __sincosf(ph, &s, &co);
  cosT[idx] = co;
  sinT[idx] = s;
}

__global__ void k_logits(const float* __restrict__ zre, const float* __restrict__ zim,
                         const float* __restrict__ cosT, const float* __restrict__ sinT,
                         float* __restrict__ logits) {
  int j = blockIdx.x;
  int tid = threadIdx.x;
  int c0 = tid, c1 = tid + 128, c2 = tid + 256;
  float a0 = 0.f, a1 = 0.f, a2 = 0.f;
  const float* zr = zre + (long)j * KLT;
  const float* zi = zim + (long)j * KLT;
  for (int kl = 0; kl < KLT; ++kl) {
    float r = zr[kl], im = zi[kl];
    const float* ct = cosT + (long)kl * CCV;
    const float* st = sinT + (long)kl * CCV;
    a0 += r * ct[c0] - im * st[c0];
    a1 += r * ct[c1] - im * st[c1];
    if (c2 < CCV) a2 += r * ct[c2] - im * st[c2];
  }
  logits[(long)j * CCV + c0] = a0;
  logits[(long)j * CCV + c1] = a1;
  if (c2 < CCV) logits[(long)j * CCV + c2] = a2;
}

__global__ void k_softmax(float* __restrict__ logits) {
  __shared__ float red[128];
  int j = blockIdx.x, tid = threadIdx.x;
  float* row = logits + (long)j * CCV;
  float m = -1e30f;
  for (int c = tid; c < CCV; c += 128) m = fmaxf(m, row[c]);
  red[tid] = m; __syncthreads();
  for (int s = 64; s > 0; s >>= 1) {
    if (tid < s) red[tid] = fmaxf(red[tid], red[tid + s]);
    __syncthreads();
  }
  m = red[0];
  __syncthreads();
  float sum = 0.f;
  for (int c = tid; c < CCV; c += 128) {
    float e = __expf(row[c] - m);
    row[c] = e;
    sum += e;
  }
  red[tid] = sum; __syncthreads();
  for (int s = 64; s > 0; s >>= 1) {
    if (tid < s) red[tid] += red[tid + s];
    __syncthreads();
  }
  float inv = 1.0f / red[0];
  for (int c = tid; c < CCV; c += 128) row[c] *= inv;
}

// Ca[d][c] = sum_j conv_w[d][j] * a[j][c], padded to 272 rows (rows>=270 zero)
__global__ void k_ca(const float* __restrict__ convw, const float* __restrict__ a,
                     float* __restrict__ ca) {
  int d = blockIdx.x, tid = threadIdx.x;
  int c0 = tid, c1 = tid + 128, c2 = tid + 256;
  float a0 = 0.f, a1 = 0.f, a2 = 0.f;
  if (d < D1V) {
    const float* w = convw + (long)d * D1V;
    for (int jj = 0; jj < D1V; ++jj) {
      float wv = w[jj];
      const float* ar = a + (long)jj * CCV;
      a0 += wv * ar[c0];
      a1 += wv * ar[c1];
      if (c2 < CCV) a2 += wv * ar[c2];
    }
  }
  ca[(long)d * CCV + c0] = a0;
  ca[(long)d * CCV + c1] = a1;
  if (c2 < CCV) ca[(long)d * CCV + c2] = a2;
}

// cvec[b][m] = (subj_w[s[b]] @ conv_b)[m] + subj_b[s[b]][m]
__global__ void k_cvec(const int* __restrict__ sidx, const float* __restrict__ subjw,
                       const float* __restrict__ subjb, const float* __restrict__ convb,
                       float* __restrict__ cvec) {
  int idx = blockIdx.x * blockDim.x + threadIdx.x;
  if (idx >= BBV * D1V) return;
  int b = idx / D1V, m = idx - b * D1V;
  int s = sidx[b];
  const float* w = subjw + ((long)s * D1V + m) * D1V;
  float acc = subjb[(long)s * D1V + m];
  for (int d = 0; d < D1V; ++d) acc += w[d] * convb[d];
  cvec[(long)b * MPAD + m] = acc;
}

// ---------------- batched WMMA fp32 GEMM ----------------
// C[b] (Mstore x N, ldc) = A[b or gather] (Mvalid x Kvalid, lda, zero-padded
// to MPAD x KDIM) * B[b] (KDIM x N, ldb) [+ bias[b][m]]
//
// B staging uses GLOBAL_LOAD_ASYNC_TO_LDS_B128 (ASYNCcnt) double-buffered;
// columns >= N may load garbage but GEMM columns are independent and stores
// are guarded, and the OOB reads stay inside the mapped workspace.
__global__ __launch_bounds__(256)
void k_gemm(const float* __restrict__ Ag, long aBatchStride,
            const int* __restrict__ gatherIdx, int lda, int Mvalid, int Kvalid,
            const float* __restrict__ Bg, long bBatchStride, int ldb, int N,
            float* __restrict__ Cg, long cBatchStride, int ldc, int Mstore,
            const float* __restrict__ bias) {
  __shared__ __align__(16) float As[MBLK * ASTR];
  __shared__ __align__(16) float Bs[2][KCH * NBLK];

  int tid = threadIdx.x;
  int lane = tid & 31;
  int wave = tid >> 5;
  int mblk = blockIdx.x;
  int nblkI = blockIdx.y;
  int b = blockIdx.z;

  const float* Ab = Ag + (gatherIdx ? (long)gatherIdx[b] : (long)b) * aBatchStride;
  const float* Bb = Bg + (long)b * bBatchStride;
  float* Cb = Cg + (long)b * cBatchStride;

  int mbase = mblk * MBLK;
  int nbase = nblkI * NBLK;

  // Per-thread B-staging coordinates: 16 rows x 128 cols, 8 floats per thread
  int brow = tid >> 4;        // 0..15
  int bcol = (tid & 15) * 8;  // 0..120
  const float* bsrc0 = Bb + (long)brow * ldb + nbase + bcol;
  float* bdst0 = &Bs[0][brow * NBLK + bcol];
  float* bdst1 = &Bs[1][brow * NBLK + bcol];

  // Kick off async staging of the first B chunk immediately
  async_copy_b128(bsrc0, bdst0);
  async_copy_b128(bsrc0 + 4, bdst0 + 4);

  // Load A block (MBLK x KDIM) into LDS with zero padding (amortized over K loop)
  for (int i = tid; i < MBLK * KDIM; i += 256) {
    int r = i / KDIM, c = i - r * KDIM;
    int gr = mbase + r;
    float v = 0.0f;
    if (gr < Mvalid && c < Kvalid) v = Ab[(long)gr * lda + c];
    As[r * ASTR + c] = v;
  }

  async_wait0();
  __syncthreads();

  v8f acc[6] = {};

  int khalf = (lane >> 4) << 1;  // 0 for lanes 0-15, 2 for lanes 16-31
  int nlane = lane & 15;
  int p = 0;
  const int NCHUNK = KDIM / KCH;  // 17

  for (int ch = 0; ch < NCHUNK; ++ch) {
    // Prefetch two chunks ahead into GL2 (global_prefetch_b8)
    if (ch + 2 < NCHUNK) {
      __builtin_prefetch(bsrc0 + (long)(ch + 2) * KCH * ldb, 0, 1);
    }
    // Issue async staging of the next chunk into the idle buffer; it lands in
    // LDS while the WMMAs below execute.
    if (ch + 1 < NCHUNK) {
      const float* src = bsrc0 + (long)(ch + 1) * KCH * ldb;
      float* dst = (p == 0) ? bdst1 : bdst0;
      async_copy_b128(src, dst);
      async_copy_b128(src + 4, dst + 4);
    }

    // Compute on buffer p: 4 K-steps x 6 m-subtiles of V_WMMA_F32_16X16X4_F32
    const float* bsp = &Bs[p][0];
    int kch0 = ch * KCH;
    #pragma unroll
    for (int kk = 0; kk < KCH; kk += 4) {
      v2f bf;
      const float* bpp = bsp + (kk + khalf) * NBLK + wave * 16 + nlane;
      bf.x = bpp[0];
      bf.y = bpp[NBLK];
      #pragma unroll
      for (int ms = 0; ms < 6; ++ms) {
        const float* app = &As[(ms * 16 + nlane) * ASTR + kch0 + kk + khalf];
        v2f af;
        af.x = app[0];
        af.y = app[1];
        acc[ms] = __builtin_amdgcn_wmma_f32_16x16x4_f32(
            false, af, false, bf, (short)0, acc[ms], false, false);
      }
    }

    if (ch + 1 < NCHUNK) async_wait0();
    __syncthreads();
    p ^= 1;
  }

  // Epilogue: scatter C fragments (+bias), guarded by true dims
  int ncol = nbase + wave * 16 + nlane;
  int moff = (lane < 16) ? 0 : 8;
  #pragma unroll
  for (int ms = 0; ms < 6; ++ms) {
    int m0 = mbase + ms * 16 + moff;
    #pragma unroll
    for (int r = 0; r < 8; ++r) {
      int m = m0 + r;
      if (m < Mstore && ncol < N) {
        float v = acc[ms][r];
        if (bias) v += bias[(long)b * MPAD + m];
        Cb[(long)m * ldc + ncol] = v;
      }
    }
  }
}

// ---------------- launcher ----------------

extern "C" void kernel_launch(void* const* d_in, const int* in_sizes, int n_in,
                              void* d_out, int out_size, void* d_ws, size_t ws_size,
                              hipStream_t stream) {
  const float* X     = (const float*)d_in[0];
  const int*   sidx  = (const int*)  d_in[1];
  const float* loc   = (const float*)d_in[2];
  const float* zre   = (const float*)d_in[3];
  const float* zim   = (const float*)d_in[4];
  const float* convw = (const float*)d_in[5];
  const float* convb = (const float*)d_in[6];
  const float* subjw = (const float*)d_in[7];
  const float* subjb = (const float*)d_in[8];
  float* out = (float*)d_out;

  float* ws     = (float*)d_ws;
  float* cosT   = ws;                  // 1024*272 = 278528
  float* sinT   = cosT + 278528;       // 278528
  float* logits = sinT + 278528;       // 270*272 = 73440 (becomes softmax a)
  float* ca     = logits + 73440;      // 272*272 = 73984 (padded conv_w@a)
  float* cvec   = ca + 73984;          // 128*288 = 36864
  float* A2     = cvec + 36864;        // 128*288*272 = 10027008

  k_trig<<<dim3((KLT * CCV + 255) / 256), dim3(256), 0, stream>>>(loc, cosT, sinT);
  k_logits<<<dim3(D1V), dim3(128), 0, stream>>>(zre, zim, cosT, sinT, logits);
  k_softmax<<<dim3(D1V), dim3(128), 0, stream>>>(logits);
  k_ca<<<dim3(CCV), dim3(128), 0, stream>>>(convw, logits, ca);
  k_cvec<<<dim3((BBV * D1V + 255) / 256), dim3(256), 0, stream>>>(sidx, subjw, subjb, convb, cvec);

  // A2[b] = subj_w[sidx[b]] @ Ca   (M=270->288, K=270->272, N=272)
  k_gemm<<<dim3(MPAD / MBLK, (CCV + NBLK - 1) / NBLK, BBV), dim3(256), 0, stream>>>(
      subjw, (long)D1V * D1V, sidx, D1V, D1V, D1V,
      ca, 0L, CCV, CCV,
      A2, (long)MPAD * KDIM, CCV, MPAD, nullptr);

  // out[b] = A2[b] @ X[b] + cvec[b]   (M=288 store 270, K=272, N=1024)
  k_gemm<<<dim3(MPAD / MBLK, TTV / NBLK, BBV), dim3(256), 0, stream>>>(
      A2, (long)MPAD * KDIM, nullptr, KDIM, MPAD, KDIM,
      X, (long)CCV * TTV, TTV, TTV,
      out, (long)D1V * TTV, TTV, D1V, cvec);
}